// DeSegaMamba_37099927503044
// MI455X (gfx1250) — compile-verified
//
#include <hip/hip_runtime.h>
#include <math.h>

// ---------------- model constants ----------------
#define BB      4
#define HH      64
#define WW      64
#define LL      4096
#define MM      16384
#define DMODEL  192
#define DINNER  384
#define DSTATE  16
#define DTRANK  12
#define NXPAD   48
#define NCHUNK  64
#define CHUNK   64

typedef float v2f __attribute__((ext_vector_type(2)));
typedef float v8f __attribute__((ext_vector_type(8)));
typedef unsigned int v4u __attribute__((ext_vector_type(4)));
typedef int v4i __attribute__((ext_vector_type(4)));
typedef int v8i __attribute__((ext_vector_type(8)));

union V8 { v8f v; float f[8]; };
union G0 { v4u v; unsigned int w[4]; };
union G1 { v8i v; unsigned int w[8]; };

#if __has_include(<hip/amd_detail/amd_gfx1250_TDM.h>)
#define TDM_SIX_ARGS 1
#endif

// ---------------- WMMA helper ----------------
// D(16x16 f32) = A(16x4 f32) x B(4x16 f32) + C
// A frag: lane(half=l>>4, idx=l&15): a.x = A[idx][2*half], a.y = A[idx][2*half+1]
// B frag: b.x = B[2*half][idx],      b.y = B[2*half+1][idx]
// C/D frag: f[v] = D[v + 8*half][idx]
__device__ __forceinline__ v8f wmma4(v2f a, v2f b, v8f c) {
  return __builtin_amdgcn_wmma_f32_16x16x4_f32(false, a, false, b, (short)0, c, false, false);
}

__device__ __forceinline__ float silu_f(float x) {
  return x / (1.0f + __expf(-x));
}

// ---------------- TDM helper: 2D tile (tile_w cols x tile_h rows, fp32) ----------------
// D# per CDNA5 ISA. tensor dims set == tile dims (no OOB),
// row stride in elements. Issued wave-level (EXEC ignored), tracked by TENSORcnt.
__device__ __forceinline__ void tdm_load_2d(unsigned int lds_off, const void* gptr,
                                            unsigned int tile_w, unsigned int tile_h,
                                            unsigned int row_stride_elems) {
  const unsigned long long ga = (unsigned long long)(size_t)gptr;
  G0 g0;
  g0.w[0] = 1u;                                   // count=1, user mode, no gather
  g0.w[1] = lds_off;                              // lds_addr (bytes)
  g0.w[2] = (unsigned int)(ga & 0xffffffffu);     // global_addr[31:0]
  g0.w[3] = (unsigned int)((ga >> 32) & 0x1ffffffu) | (2u << 30); // addr[56:32] | type=2
  G1 g1;
  g1.w[0] = 2u << 16;                             // workgroup_mask=0, data_size=4B
  g1.w[1] = (tile_w & 0xffffu) << 16;             // tensor_dim0[15:0] @bit48
  g1.w[2] = ((tile_w >> 16) & 0xffffu) | ((tile_h & 0xffffu) << 16);  // dim0 hi | dim1 lo
  g1.w[3] = ((tile_h >> 16) & 0xffffu) | ((tile_w & 0xffffu) << 16);  // dim1 hi | tile_dim0
  g1.w[4] = tile_h & 0xffffu;                     // tile_dim1 (tile_dim2=0 -> 2D)
  g1.w[5] = row_stride_elems;                     // tensor_dim0_stride[31:0]
  g1.w[6] = 0u;                                   // stride hi, dim1_stride lo
  g1.w[7] = 0u;
  v4i z4 = {0, 0, 0, 0};
#ifdef TDM_SIX_ARGS
  v8i z8 = {0, 0, 0, 0, 0, 0, 0, 0};
  __builtin_amdgcn_tensor_load_to_lds(g0.v, g1.v, z4, z4, z8, 0);
#else
  __builtin_amdgcn_tensor_load_to_lds(g0.v, g1.v, z4, z4, 0);
#endif
}

__device__ __forceinline__ unsigned int lds_off_of(const void* shared_ptr) {
  // ISA §10.2: flat LDS address low 32 bits == LDS byte offset
  return (unsigned int)(size_t)shared_ptr;
}

// ---------------- K1: xz = x @ W_in ; split -> xi, z ----------------
__global__ void k_inproj(const float* __restrict__ x, const float* __restrict__ Win,
                         float* __restrict__ xi, float* __restrict__ z) {
  const int wave = threadIdx.x >> 5, lane = threadIdx.x & 31;
  const int half = lane >> 4, idx = lane & 15;
  const int m0 = blockIdx.x * 16;
  const int col = blockIdx.y * 64 + wave * 16 + idx;
  V8 acc; acc.v = (v8f){};
  const float* arow = x + (size_t)(m0 + idx) * DMODEL;
  for (int k = 0; k < DMODEL; k += 4) {
    const int kr = k + 2 * half;
    v2f a = *(const v2f*)(arow + kr);
    v2f b; b.x = Win[(size_t)kr * 768 + col]; b.y = Win[(size_t)(kr + 1) * 768 + col];
    acc.v = wmma4(a, b, acc.v);
  }
  float* dst = (col < DINNER) ? xi : z;
  const int cc = (col < DINNER) ? col : (col - DINNER);
  for (int v = 0; v < 8; ++v)
    dst[(size_t)(m0 + v + 8 * half) * DINNER + cc] = acc.f[v];
}

// ---------------- K2: depthwise 3x3 conv + bias + SiLU (NHWC) ----------------
__global__ void k_dwconv(const float* __restrict__ xi, const float* __restrict__ cw,
                         const float* __restrict__ cb, float* __restrict__ u) {
  const int c = threadIdx.x;
  const int hw = blockIdx.x, b = blockIdx.y;
  const int h = hw >> 6, w = hw & 63;
  float acc = cb[c];
  #pragma unroll
  for (int dy = 0; dy < 3; ++dy) {
    const int hh = h + dy - 1;
    if (hh < 0 || hh >= HH) continue;
    #pragma unroll
    for (int dx = 0; dx < 3; ++dx) {
      const int wv = w + dx - 1;
      if (wv < 0 || wv >= WW) continue;
      acc = fmaf(xi[((size_t)b * LL + hh * WW + wv) * DINNER + c],
                 cw[c * 9 + dy * 3 + dx], acc);
    }
  }
  u[((size_t)b * LL + hw) * DINNER + c] = silu_f(acc);
}

// ---------------- K3a: xdbl = u @ Wx (N=44 padded to 48) ----------------
__global__ void k_xdbl(const float* __restrict__ u, const float* __restrict__ Wx,
                       float* __restrict__ xdbl) {
  const int wave = threadIdx.x >> 5, lane = threadIdx.x & 31;
  const int half = lane >> 4, idx = lane & 15;
  const int m0 = blockIdx.x * 16;
  const int col = wave * 16 + idx;
  const bool ok = (col < DTRANK + 2 * DSTATE);
  V8 acc; acc.v = (v8f){};
  const float* arow = u + (size_t)(m0 + idx) * DINNER;
  for (int k = 0; k < DINNER; k += 4) {
    const int kr = k + 2 * half;
    v2f a = *(const v2f*)(arow + kr);
    v2f b;
    b.x = ok ? Wx[(size_t)kr * 44 + col] : 0.0f;
    b.y = ok ? Wx[(size_t)(kr + 1) * 44 + col] : 0.0f;
    acc.v = wmma4(a, b, acc.v);
  }
  for (int v = 0; v < 8; ++v)
    xdbl[(size_t)(m0 + v + 8 * half) * NXPAD + col] = acc.f[v];
}

// ---------------- K3b: Cs = xdbl[:,28:44] + prompt @ Wp ----------------
__global__ void k_cs(const float* __restrict__ prompt, const float* __restrict__ Wp,
                     const float* __restrict__ xdbl, float* __restrict__ Cs) {
  const int wave = threadIdx.x >> 5, lane = threadIdx.x & 31;
  const int half = lane >> 4, idx = lane & 15;
  const int m0 = (blockIdx.x * 4 + wave) * 16;
  V8 acc;
  for (int v = 0; v < 8; ++v)
    acc.f[v] = xdbl[(size_t)(m0 + v + 8 * half) * NXPAD + (DTRANK + DSTATE) + idx];
  const float* arow = prompt + (size_t)(m0 + idx) * DMODEL;
  for (int k = 0; k < DMODEL; k += 4) {
    const int kr = k + 2 * half;
    v2f a = *(const v2f*)(arow + kr);
    v2f b; b.x = Wp[kr * DSTATE + idx]; b.y = Wp[(kr + 1) * DSTATE + idx];
    acc.v = wmma4(a, b, acc.v);
  }
  for (int v = 0; v < 8; ++v)
    Cs[(size_t)(m0 + v + 8 * half) * DSTATE + idx] = acc.f[v];
}

// ---------------- K3c: delta = softplus(dtr @ Wdt + b_dt) ----------------
__global__ void k_delta(const float* __restrict__ xdbl, const float* __restrict__ Wdt,
                        const float* __restrict__ b_dt, float* __restrict__ delta) {
  const int wave = threadIdx.x >> 5, lane = threadIdx.x & 31;
  const int half = lane >> 4, idx = lane & 15;
  const int m0 = blockIdx.x * 16;
  const int col = (blockIdx.y * 4 + wave) * 16 + idx;
  V8 acc; acc.v = (v8f){};
  const float* arow = xdbl + (size_t)(m0 + idx) * NXPAD;   // dtr = cols 0..11
  for (int k = 0; k < DTRANK; k += 4) {
    const int kr = k + 2 * half;
    v2f a = *(const v2f*)(arow + kr);
    v2f b; b.x = Wdt[kr * DINNER + col]; b.y = Wdt[(kr + 1) * DINNER + col];
    acc.v = wmma4(a, b, acc.v);
  }
  const float bd = b_dt[col];
  for (int v = 0; v < 8; ++v) {
    float t = acc.f[v] + bd;
    t = (t > 20.0f) ? t : log1pf(__expf(t));   // softplus
    delta[(size_t)(m0 + v + 8 * half) * DINNER + col] = t;
  }
}

// ---------------- K4: chunked selective scan, TDM-staged tiles ----------------
// Each block handles (chunk c, 16 channels d, batch b). Wave 0 issues TDM DMAs of the
// strided 2D tiles (64 rows x 16 cols) into LDS; inner loop is pure LDS + VALU.
// pass1: local scan with h0=0, record end state h and transition product P.
__global__ void k_scan_pass1(const float* __restrict__ delta, const float* __restrict__ u,
                             const float* __restrict__ xdbl, const float* __restrict__ A_log,
                             float* __restrict__ hend, float* __restrict__ Pend) {
  __shared__ float sh_d[CHUNK * 16];
  __shared__ float sh_u[CHUNK * 16];
  __shared__ float sh_b[CHUNK * 16];
  const int n  = threadIdx.x & 15;
  const int dl = threadIdx.x >> 4;
  const int d0 = blockIdx.y * 16;
  const int d  = d0 + dl;
  const int b  = blockIdx.z;
  const int c  = blockIdx.x;
  const size_t row0 = (size_t)b * LL + (size_t)c * CHUNK;

  if (threadIdx.x < 32) {   // wave 0 issues the tile DMAs
    tdm_load_2d(lds_off_of(sh_d), &delta[row0 * DINNER + d0], 16, CHUNK, DINNER);
    tdm_load_2d(lds_off_of(sh_u), &u[row0 * DINNER + d0],     16, CHUNK, DINNER);
    tdm_load_2d(lds_off_of(sh_b), &xdbl[row0 * NXPAD + DTRANK], 16, CHUNK, NXPAD);
    __builtin_amdgcn_s_wait_tensorcnt(0);
  }
  __syncthreads();

  const float Adn = -__expf(A_log[d * DSTATE + n]);
  float h = 0.0f, P = 1.0f;
  for (int i = 0; i < CHUNK; ++i) {
    const float dlt = sh_d[i * 16 + dl];
    const float uu  = sh_u[i * 16 + dl];
    const float bsv = sh_b[i * 16 + n];
    const float dA  = __expf(dlt * Adn);
    h = fmaf(dA, h, dlt * uu * bsv);
    P *= dA;
  }
  const size_t o = ((size_t)c * BB + b) * (DINNER * DSTATE) + (size_t)d * DSTATE + n;
  hend[o] = h; Pend[o] = P;
}

// pass2: sequential combine over chunks (depth NCHUNK, width B*DINNER*DSTATE)
__global__ void k_scan_combine(const float* __restrict__ hend, const float* __restrict__ Pend,
                               float* __restrict__ hstart) {
  const int idx = blockIdx.x * blockDim.x + threadIdx.x;  // 0 .. 24575
  float h = 0.0f;
  for (int c = 0; c < NCHUNK; ++c) {
    const size_t o = (size_t)c * (BB * DINNER * DSTATE) + idx;
    hstart[o] = h;
    h = fmaf(Pend[o], h, hend[o]);
  }
}

// pass3: re-run each chunk from its true start state, emit y = h.C + u*D
__global__ void k_scan_pass3(const float* __restrict__ delta, const float* __restrict__ u,
                             const float* __restrict__ xdbl, const float* __restrict__ Cs,
                             const float* __restrict__ A_log, const float* __restrict__ Dvec,
                             const float* __restrict__ hstart, float* __restrict__ y) {
  __shared__ float sh_d[CHUNK * 16];
  __shared__ float sh_u[CHUNK * 16];
  __shared__ float sh_b[CHUNK * 16];
  __shared__ float sh_c[CHUNK * 16];
  const int n  = threadIdx.x & 15;
  const int dl = threadIdx.x >> 4;
  const int d0 = blockIdx.y * 16;
  const int d  = d0 + dl;
  const int b  = blockIdx.z;
  const int c  = blockIdx.x;
  const size_t row0 = (size_t)b * LL + (size_t)c * CHUNK;

  if (threadIdx.x < 32) {
    tdm_load_2d(lds_off_of(sh_d), &delta[row0 * DINNER + d0], 16, CHUNK, DINNER);
    tdm_load_2d(lds_off_of(sh_u), &u[row0 * DINNER + d0],     16, CHUNK, DINNER);
    tdm_load_2d(lds_off_of(sh_b), &xdbl[row0 * NXPAD + DTRANK], 16, CHUNK, NXPAD);
    tdm_load_2d(lds_off_of(sh_c), &Cs[row0 * DSTATE],         16, CHUNK, DSTATE);
    __builtin_amdgcn_s_wait_tensorcnt(0);
  }
  __syncthreads();

  const float Adn = -__expf(A_log[d * DSTATE + n]);
  const float Dd  = Dvec[d];
  float h = hstart[((size_t)c * BB + b) * (DINNER * DSTATE) + (size_t)d * DSTATE + n];
  size_t row = row0;
  for (int i = 0; i < CHUNK; ++i, ++row) {
    const float dlt = sh_d[i * 16 + dl];
    const float uu  = sh_u[i * 16 + dl];
    const float bsv = sh_b[i * 16 + n];
    const float csv = sh_c[i * 16 + n];
    const float dA  = __expf(dlt * Adn);
    h = fmaf(dA, h, dlt * uu * bsv);
    float part = h * csv;
    part += __shfl_xor(part, 1, 32);
    part += __shfl_xor(part, 2, 32);
    part += __shfl_xor(part, 4, 32);
    part += __shfl_xor(part, 8, 32);
    if (n == 0) y[row * DINNER + d] = fmaf(uu, Dd, part);
  }
}

// ---------------- K5: LayerNorm(384) + gate with silu(z) ----------------
__global__ void k_lngate(const float* __restrict__ y, const float* __restrict__ z,
                         const float* __restrict__ ln_g, const float* __restrict__ ln_b,
                         float* __restrict__ yg) {
  const int row = blockIdx.x;
  const float* yr = y + (size_t)row * DINNER;
  const float* zr = z + (size_t)row * DINNER;
  const int t = threadIdx.x;
  float v0 = yr[t], v1 = yr[t + 128], v2 = yr[t + 256];
  float s  = v0 + v1 + v2;
  float s2 = v0 * v0 + v1 * v1 + v2 * v2;
  #pragma unroll
  for (int m = 1; m < 32; m <<= 1) {
    s  += __shfl_xor(s, m, 32);
    s2 += __shfl_xor(s2, m, 32);
  }
  __shared__ float ss[4], ss2[4];
  const int wave = t >> 5, lane = t & 31;
  if (lane == 0) { ss[wave] = s; ss2[wave] = s2; }
  __syncthreads();
  s  = ss[0] + ss[1] + ss[2] + ss[3];
  s2 = ss2[0] + ss2[1] + ss2[2] + ss2[3];
  const float mu   = s * (1.0f / DINNER);
  const float var  = s2 * (1.0f / DINNER) - mu * mu;
  const float rstd = rsqrtf(var + 1e-5f);
  #pragma unroll
  for (int j = 0; j < 3; ++j) {
    const int c = t + j * 128;
    const float val = (yr[c] - mu) * rstd * ln_g[c] + ln_b[c];
    const float zc  = zr[c];
    yg[(size_t)row * DINNER + c] = val * silu_f(zc);
  }
}

// ---------------- K6: out = yg @ Wout ----------------
__global__ void k_out(const float* __restrict__ yg, const float* __restrict__ Wout,
                      float* __restrict__ out) {
  const int wave = threadIdx.x >> 5, lane = threadIdx.x & 31;
  const int half = lane >> 4, idx = lane & 15;
  const int m0 = blockIdx.x * 16;
  const int col = blockIdx.y * 64 + wave * 16 + idx;
  V8 acc; acc.v = (v8f){};
  const float* arow = yg + (size_t)(m0 + idx) * DINNER;
  for (int k = 0; k < DINNER; k += 4) {
    const int kr = k + 2 * half;
    v2f a = *(const v2f*)(arow + kr);
    v2f b; b.x = Wout[kr * DMODEL + col]; b.y = Wout[(kr + 1) * DMODEL + col];
    acc.v = wmma4(a, b, acc.v);
  }
  for (int v = 0; v < 8; ++v)
    out[(size_t)(m0 + v + 8 * half) * DMODEL + col] = acc.f[v];
}

// ---------------- launcher ----------------
extern "C" void kernel_launch(void* const* d_in, const int* in_sizes, int n_in,
                              void* d_out, int out_size, void* d_ws, size_t ws_size,
                              hipStream_t stream) {
  (void)in_sizes; (void)n_in; (void)out_size; (void)ws_size;
  const float* x      = (const float*)d_in[0];
  const float* prompt = (const float*)d_in[1];
  const float* W_in   = (const float*)d_in[2];
  const float* conv_w = (const float*)d_in[3];
  const float* conv_b = (const float*)d_in[4];
  const float* Wx     = (const float*)d_in[5];
  const float* Wdt    = (const float*)d_in[6];
  const float* b_dt   = (const float*)d_in[7];
  const float* A_log  = (const float*)d_in[8];
  const float* Dvec   = (const float*)d_in[9];
  const float* Wp     = (const float*)d_in[10];
  const float* ln_g   = (const float*)d_in[11];
  const float* ln_b   = (const float*)d_in[12];
  const float* Wout   = (const float*)d_in[13];
  float* out = (float*)d_out;
  float* ws  = (float*)d_ws;

  // workspace layout (floats); xi reused for delta, u reused for yg
  float* bufXI = ws;                               // MM*DINNER (xi, then delta)
  float* bufZ  = bufXI + (size_t)MM * DINNER;      // MM*DINNER
  float* bufU  = bufZ  + (size_t)MM * DINNER;      // MM*DINNER (u, then yg)
  float* bufY  = bufU  + (size_t)MM * DINNER;      // MM*DINNER
  float* xdbl  = bufY  + (size_t)MM * DINNER;      // MM*NXPAD
  float* Cs    = xdbl  + (size_t)MM * NXPAD;       // MM*DSTATE
  float* hend  = Cs    + (size_t)MM * DSTATE;      // NCHUNK*B*DINNER*DSTATE
  float* Pend  = hend  + (size_t)NCHUNK * BB * DINNER * DSTATE;
  float* hst   = Pend  + (size_t)NCHUNK * BB * DINNER * DSTATE;

  k_inproj<<<dim3(MM / 16, (2 * DINNER) / 64), 128, 0, stream>>>(x, W_in, bufXI, bufZ);
  k_dwconv<<<dim3(HH * WW, BB), DINNER, 0, stream>>>(bufXI, conv_w, conv_b, bufU);
  k_xdbl<<<dim3(MM / 16), 96, 0, stream>>>(bufU, Wx, xdbl);
  k_cs<<<dim3(MM / 64), 128, 0, stream>>>(prompt, Wp, xdbl, Cs);
  k_delta<<<dim3(MM / 16, DINNER / 64), 128, 0, stream>>>(xdbl, Wdt, b_dt, bufXI);
  dim3 gs(NCHUNK, DINNER / 16, BB);
  k_scan_pass1<<<gs, 256, 0, stream>>>(bufXI, bufU, xdbl, A_log, hend, Pend);
  k_scan_combine<<<dim3((BB * DINNER * DSTATE) / 256), 256, 0, stream>>>(hend, Pend, hst);
  k_scan_pass3<<<gs, 256, 0, stream>>>(bufXI, bufU, xdbl, Cs, A_log, Dvec, hst, bufY);
  k_lngate<<<dim3(MM), 128, 0, stream>>>(bufY, bufZ, ln_g, ln_b, bufU);
  k_out<<<dim3(MM / 16, DMODEL / 64), 128, 0, stream>>>(bufU, Wout, out);
}